// GeneAttentionLayer_16810501996741
// MI455X (gfx1250) — compile-verified
//
#include <hip/hip_runtime.h>
#include <hip/hip_bf16.h>
#include <float.h>

#define N_NODES   50000
#define EMBED_DIM 128
#define N_EDGES   600000
#define ALPHA     0.2f

typedef __attribute__((ext_vector_type(2))) float v2f;
typedef __attribute__((ext_vector_type(8))) float v8f;

// Monotone float <-> uint encoding so atomicMax(unsigned) gives float max.
__device__ __forceinline__ unsigned encodeOrderedF32(float f) {
    unsigned u = __float_as_uint(f);
    return (u & 0x80000000u) ? ~u : (u | 0x80000000u);
}
__device__ __forceinline__ float decodeOrderedF32(unsigned u) {
    u = (u & 0x80000000u) ? (u & 0x7FFFFFFFu) : ~u;
    return __uint_as_float(u);
}

__global__ void k_init(unsigned* __restrict__ maxSlot, float* __restrict__ sumSlot) {
    if (threadIdx.x == 0) {
        maxSlot[0] = encodeOrderedF32(-FLT_MAX);
        sumSlot[0] = 0.0f;
    }
}

// out = emb  (so the scatter pass lands on top of the residual)
__global__ void k_copy(const float* __restrict__ emb, float* __restrict__ out) {
    int i = blockIdx.x * blockDim.x + threadIdx.x;
    if (i < (N_NODES * EMBED_DIM) / 4) {
        ((float4*)out)[i] = ((const float4*)emb)[i];
    }
}

// One wave handles 16 edges. score[m] = leaky_relu(dot(emb[src[m]], emb[dst[m]])).
// The 16 dots are the diagonal of a 16x16 WMMA product, accumulated over K=128
// in chunks of 4 via V_WMMA_F32_16X16X4_F32.
// Launch: gridDim = 9375, blockDim = 128 (4 waves) -> exactly 37500 tiles,
// so every wave is fully active (EXEC all-1s around the WMMAs).
__global__ void k_scores(const float* __restrict__ emb,
                         const int*   __restrict__ rel,
                         float*       __restrict__ scores,
                         unsigned*    __restrict__ maxSlot) {
    __shared__ float smax[4];
    const int lane = threadIdx.x & 31;
    const int wave = threadIdx.x >> 5;
    const int tile = blockIdx.x * 4 + wave;          // < 37500 always

    // A 16x4 f32 layout: lane L holds row m=L&15, K-pair selected by L>>4.
    // B 4x16 f32 layout: lane L holds col n=L&15, same K-pair. Diagonal needs
    // m == n == edge index, so both rows come from the same edge per lane.
    const int m     = lane & 15;
    const int khalf = lane >> 4;                     // 0 or 1
    const int e     = tile * 16 + m;
    const int si    = rel[2 * e + 0];
    const int di    = rel[2 * e + 1];
    const float* rowI = emb + (long long)si * EMBED_DIM;   // A source (drug_i)
    const float* rowJ = emb + (long long)di * EMBED_DIM;   // B source (drug_j)

    v8f c = {};
#pragma unroll 8
    for (int kc = 0; kc < 32; ++kc) {
        const int off = kc * 4 + khalf * 2;
        v2f a = *(const v2f*)(rowI + off);
        v2f b = *(const v2f*)(rowJ + off);
        // D = A(16x4) * B(4x16) + C, f32 — emits v_wmma_f32_16x16x4_f32
        c = __builtin_amdgcn_wmma_f32_16x16x4_f32(
                /*neg_a=*/false, a, /*neg_b=*/false, b,
                /*c_mod=*/(short)0, c, /*reuse_a=*/false, /*reuse_b=*/false);
    }

    // Diagonal D[m][m]: lanes 0..7 hold m=0..7 in c[m]; lanes 24..31 hold
    // m=8..15 in c[lane-24].  idx = lane & 7 in both cases.
    const int idx = lane & 7;
    float d01 = (idx == 0) ? c[0] : c[1];
    float d23 = (idx == 2) ? c[2] : c[3];
    float d45 = (idx == 4) ? c[4] : c[5];
    float d67 = (idx == 6) ? c[6] : c[7];
    float dlo = (idx < 2) ? d01 : d23;
    float dhi = (idx < 6) ? d45 : d67;
    float diag = (idx < 4) ? dlo : dhi;

    float myScore = -FLT_MAX;
    const bool active = (lane < 8) || (lane >= 24);
    if (active) {
        const int em = (lane < 8) ? lane : (lane - 16);
        float s = (diag >= 0.0f) ? diag : ALPHA * diag;   // leaky_relu
        scores[tile * 16 + em] = s;
        myScore = s;
    }

    // Block max -> global ordered atomicMax (for softmax stability)
    for (int off = 16; off > 0; off >>= 1)
        myScore = fmaxf(myScore, __shfl_xor(myScore, off, 32));
    if (lane == 0) smax[wave] = myScore;
    __syncthreads();
    if (wave == 0) {
        float v = (lane < 4) ? smax[lane] : -FLT_MAX;
        for (int off = 16; off > 0; off >>= 1)
            v = fmaxf(v, __shfl_xor(v, off, 32));
        if (lane == 0) atomicMax(maxSlot, encodeOrderedF32(v));
    }
}

// scores[e] <- exp(scores[e] - max); accumulate global sum.
__global__ void k_expsum(float* __restrict__ scores,
                         const unsigned* __restrict__ maxSlot,
                         float* __restrict__ sumSlot) {
    __shared__ float ssum[8];
    const int lane = threadIdx.x & 31;
    const int wave = threadIdx.x >> 5;
    const int e = blockIdx.x * blockDim.x + threadIdx.x;
    const float mx = decodeOrderedF32(*maxSlot);
    float v = 0.0f;
    if (e < N_EDGES) {
        v = __expf(scores[e] - mx);
        scores[e] = v;
    }
    for (int off = 16; off > 0; off >>= 1)
        v += __shfl_xor(v, off, 32);
    if (lane == 0) ssum[wave] = v;
    __syncthreads();
    if (wave == 0) {
        float t = (lane < (int)(blockDim.x >> 5)) ? ssum[lane] : 0.0f;
        for (int off = 16; off > 0; off >>= 1)
            t += __shfl_xor(t, off, 32);
        if (lane == 0) atomicAdd(sumSlot, t);
    }
}

// One wave per edge; each lane owns 4 dims. out[src] += (att/sum) * emb[dst].
__global__ void k_scatter(const float* __restrict__ emb,
                          const int*   __restrict__ rel,
                          const float* __restrict__ att,
                          const float* __restrict__ sumSlot,
                          float*       __restrict__ out) {
    const int lane = threadIdx.x & 31;
    const int wave = threadIdx.x >> 5;
    const long long e = (long long)blockIdx.x * (blockDim.x >> 5) + wave;
    if (e >= N_EDGES) return;                         // wave-uniform guard

    const float w  = att[e] * (1.0f / *sumSlot);
    const int  si  = rel[2 * e + 0];
    const int  di  = rel[2 * e + 1];
    const float4 vj = *(const float4*)(emb + (long long)di * EMBED_DIM + lane * 4);
    float* o = out + (long long)si * EMBED_DIM + lane * 4;
    atomicAdd(o + 0, w * vj.x);
    atomicAdd(o + 1, w * vj.y);
    atomicAdd(o + 2, w * vj.z);
    atomicAdd(o + 3, w * vj.w);
}

extern "C" void kernel_launch(void* const* d_in, const int* in_sizes, int n_in,
                              void* d_out, int out_size, void* d_ws, size_t ws_size,
                              hipStream_t stream) {
    const float* emb = (const float*)d_in[0];   // (50000, 128) f32
    const int*   rel = (const int*)d_in[1];     // (600000, 2) indices
    float* out = (float*)d_out;                 // (50000, 128) f32

    unsigned* maxSlot = (unsigned*)d_ws;
    float*    sumSlot = (float*)d_ws + 1;
    float*    scores  = (float*)d_ws + 16;      // 64B-aligned, 600000 floats

    k_init<<<1, 32, 0, stream>>>(maxSlot, sumSlot);
    k_copy<<<(N_NODES * EMBED_DIM / 4 + 255) / 256, 256, 0, stream>>>(emb, out);
    k_scores<<<(N_EDGES / 16) / 4, 128, 0, stream>>>(emb, rel, scores, maxSlot);
    k_expsum<<<(N_EDGES + 255) / 256, 256, 0, stream>>>(scores, maxSlot, sumSlot);
    k_scatter<<<(N_EDGES + 7) / 8, 256, 0, stream>>>(emb, rel, scores, sumSlot, out);
}